// GCNII_18038862643739
// MI455X (gfx1250) — compile-verified
//
#include <hip/hip_runtime.h>
#include <cmath>

typedef __attribute__((ext_vector_type(2))) float v2f;
typedef __attribute__((ext_vector_type(8))) float v8f;

#define N_NODES 50000
#define N_EDGES 800000
#define HID     128
#define OUT_C   64
#define N_LAYERS 8

// ---------------------------------------------------------------------------
// GEMM: Y = A(N x 128) @ W(128 x M), M compile-time (128 or 64).
//   COMBINE:  Y = relu(cT * A + cW * (A@W))   (requires M == 128, T == A)
// One block = 16 output rows; 8 waves each own a 16-col tile.
// A tile (16x128) and W (128xM, transposed) both staged in LDS; inner loop is
// pure ds_load_b64 + V_WMMA_F32_16X16X4_F32.
// ---------------------------------------------------------------------------
template <int M, bool COMBINE>
__global__ void __launch_bounds__(256)
gemm_wmma_f32(const float* __restrict__ A, const float* __restrict__ W,
              float* __restrict__ Y, float cT, float cW)
{
    __shared__ float As[16 * 132];        // A tile, +4 pad kills bank conflicts
    __shared__ float Wt[M * 132];         // W transposed: Wt[m][k]
    const int tid  = threadIdx.x;
    const int wave = tid >> 5;
    const int lane = tid & 31;
    const int rTile = blockIdx.x;         // 50000/16 = 3125 exact

    // Stage 16x128 A tile (coalesced float4 -> vector LDS store)
#pragma unroll
    for (int i = tid; i < 512; i += 256) {
        int r = (i << 2) >> 7;
        int c = (i << 2) & 127;
        float4 a4 = *reinterpret_cast<const float4*>(
            A + ((unsigned)rTile * 16 + r) * 128 + c);
        *reinterpret_cast<float4*>(&As[r * 132 + c]) = a4;
    }
    // Stage W transposed: read W[k][m..m+3] coalesced, scatter to Wt[m][k]
#pragma unroll
    for (int i = tid; i < (128 * M) / 4; i += 256) {
        int k = (i << 2) / M;             // M power of two -> shift
        int m = (i << 2) & (M - 1);
        float4 w4 = *reinterpret_cast<const float4*>(W + k * M + m);
        Wt[(m + 0) * 132 + k] = w4.x;
        Wt[(m + 1) * 132 + k] = w4.y;
        Wt[(m + 2) * 132 + k] = w4.z;
        Wt[(m + 3) * 132 + k] = w4.w;
    }
    __syncthreads();

    const int cBase = wave * 16;
    if (cBase >= M) return;               // wave-uniform; EXEC stays all-ones

    const int row = lane & 15;            // A-matrix: M(row) = lane % 16
    const int kk  = (lane >> 4) * 2;      // A/B VGPR v holds K = 2*(lane/16)+v
    const int col = cBase + (lane & 15);  // B/C/D: N = lane % 16

    v8f acc = {};
#pragma unroll
    for (int k0 = 0; k0 < 128; k0 += 4) {
        v2f a, b;
        a.x = As[row * 132 + k0 + kk];
        a.y = As[row * 132 + k0 + kk + 1];
        b.x = Wt[col * 132 + k0 + kk];
        b.y = Wt[col * 132 + k0 + kk + 1];
        acc = __builtin_amdgcn_wmma_f32_16x16x4_f32(
            /*neg_a=*/false, a, /*neg_b=*/false, b,
            /*c_mod=*/(short)0, acc, /*reuse_a=*/false, /*reuse_b=*/false);
    }

    // C/D layout: VGPR v, lane L -> row = 8*(L>>4)+v, col = cBase + (L&15)
    float* Yb = Y + (unsigned)rTile * 16 * M;
    const int rLoc = (lane >> 4) * 8;
#pragma unroll
    for (int v = 0; v < 8; ++v) {
        float y = acc[v];
        if (COMBINE) {
            float t = As[(rLoc + v) * 132 + col];   // T == A, tile is in LDS
            y = cT * t + cW * y;
            y = y > 0.f ? y : 0.f;
        }
        Yb[(rLoc + v) * M + col] = y;
    }
}

// ---------------------------------------------------------------------------
// SpMM scatter: out[dst] += w * feats[src], C channels, float4 per lane,
// hardware fp32 atomics (feature matrix is 25.6MB -> L2-resident).
// tpeShift: log2(C/4) threads per edge. All indices fit in 32 bits.
// ---------------------------------------------------------------------------
__global__ void __launch_bounds__(256)
spmm_scatter(const float* __restrict__ feats, const int* __restrict__ ei,
             const float* __restrict__ ew, float* __restrict__ out,
             int nEdges, int C, int tpeShift)
{
    unsigned gid = blockIdx.x * 256u + threadIdx.x;
    unsigned e   = gid >> tpeShift;
    if (e >= (unsigned)nEdges) return;
    unsigned c4 = (gid & ((1u << tpeShift) - 1u)) << 2;
    int s = ei[e];
    int d = ei[(unsigned)nEdges + e];
    float w = ew[e];
    const float4 v = *reinterpret_cast<const float4*>(feats + (unsigned)s * C + c4);
    float* o = out + (unsigned)d * C + c4;
    unsafeAtomicAdd(o + 0, w * v.x);
    unsafeAtomicAdd(o + 1, w * v.y);
    unsafeAtomicAdd(o + 2, w * v.z);
    unsafeAtomicAdd(o + 3, w * v.w);
}

__global__ void __launch_bounds__(256)
fill_zero(float* __restrict__ p, unsigned n4)    // n4 = n/4
{
    unsigned i = blockIdx.x * 256u + threadIdx.x;
    if (i < n4) reinterpret_cast<float4*>(p)[i] = float4{0.f, 0.f, 0.f, 0.f};
}

__global__ void __launch_bounds__(256)
bias_relu(const float* __restrict__ agg, const float* __restrict__ b,
          float* __restrict__ out, unsigned n, int Cmask)
{
    unsigned i = blockIdx.x * 256u + threadIdx.x;
    if (i >= n) return;
    float v = agg[i] + b[i & Cmask];
    out[i] = v > 0.f ? v : 0.f;
}

__global__ void __launch_bounds__(256)
axpby_inplace(float* __restrict__ a, const float* __restrict__ x0,
              float c0, float c1, unsigned n)
{
    unsigned i = blockIdx.x * 256u + threadIdx.x;
    if (i < n) a[i] = c0 * a[i] + c1 * x0[i];
}

__global__ void __launch_bounds__(256)
bias_add(float* __restrict__ out, const float* __restrict__ b,
         unsigned n, int Cmask)
{
    unsigned i = blockIdx.x * 256u + threadIdx.x;
    if (i < n) out[i] += b[i & Cmask];
}

// ---------------------------------------------------------------------------
extern "C" void kernel_launch(void* const* d_in, const int* in_sizes, int n_in,
                              void* d_out, int out_size, void* d_ws, size_t ws_size,
                              hipStream_t stream)
{
    const float* x        = (const float*)d_in[0];
    const int*   ei       = (const int*)  d_in[1];
    const float* ew       = (const float*)d_in[2];
    const float* W_in     = (const float*)d_in[3];
    const float* b_in     = (const float*)d_in[4];
    const float* W_layers = (const float*)d_in[5];
    const float* W_out    = (const float*)d_in[6];
    const float* b_out    = (const float*)d_in[7];
    float* out = (float*)d_out;

    const unsigned nh = (unsigned)N_NODES * HID;     // 6.4M floats
    float* bufH   = (float*)d_ws;                    // current h
    float* bufX0  = bufH  + nh;                      // x0 (h after input layer)
    float* bufAgg = bufX0 + nh;                      // spmm target / t

    dim3 blk(256);
    const int gemmGrid = N_NODES / 16;                                 // 3125
    const int fillG128 = (int)((nh / 4 + 255) / 256);
    const int ewG128   = (int)((nh + 255) / 256);
    const int spmmG128 = (int)(((long long)N_EDGES * 32 + 255) / 256);
    const int spmmG64  = (int)(((long long)N_EDGES * 16 + 255) / 256);

    // 1) xw = x @ W_in  -> bufH
    gemm_wmma_f32<HID, false><<<gemmGrid, blk, 0, stream>>>(x, W_in, bufH, 0.f, 0.f);

    // 2) agg = spmm(xw);  h0 = relu(agg + b_in) -> bufX0
    fill_zero<<<fillG128, blk, 0, stream>>>(bufAgg, nh / 4);
    spmm_scatter<<<spmmG128, blk, 0, stream>>>(bufH, ei, ew, bufAgg, N_EDGES, HID, 5);
    bias_relu<<<ewG128, blk, 0, stream>>>(bufAgg, b_in, bufX0, nh, HID - 1);

    // 3) GCNII layers
    const float* cur = bufX0;
    for (int l = 0; l < N_LAYERS; ++l) {
        float beta = logf(1.0f / (float)(l + 1) + 1.0f);
        fill_zero<<<fillG128, blk, 0, stream>>>(bufAgg, nh / 4);
        spmm_scatter<<<spmmG128, blk, 0, stream>>>(cur, ei, ew, bufAgg, N_EDGES, HID, 5);
        // t = 0.5*agg + 0.5*x0   (in place in bufAgg)
        axpby_inplace<<<ewG128, blk, 0, stream>>>(bufAgg, bufX0, 0.5f, 0.5f, nh);
        // h = relu((1-beta)*t + beta*(t @ W_l))  -> bufH (cur no longer read)
        gemm_wmma_f32<HID, true><<<gemmGrid, blk, 0, stream>>>(
            bufAgg, W_layers + (size_t)l * HID * HID, bufH, 1.0f - beta, beta);
        cur = bufH;
    }

    // 4) hw = h @ W_out -> bufX0 (x0 dead now);  out = spmm(hw) + b_out
    gemm_wmma_f32<OUT_C, false><<<gemmGrid, blk, 0, stream>>>(cur, W_out, bufX0, 0.f, 0.f);
    const unsigned no = (unsigned)N_NODES * OUT_C;
    fill_zero<<<(int)((no / 4 + 255) / 256), blk, 0, stream>>>(out, no / 4);
    spmm_scatter<<<spmmG64, blk, 0, stream>>>(bufX0, ei, ew, out, N_EDGES, OUT_C, 4);
    bias_add<<<(int)((no + 255) / 256), blk, 0, stream>>>(out, b_out, no, OUT_C - 1);
}